// Sobel_16724602651101
// MI455X (gfx1250) — compile-verified
//
#include <hip/hip_runtime.h>
#include <stdint.h>

// Problem constants (reference: x = (16, 3, 512, 512) f32)
#define CH 3
#define IMG_H 512
#define IMG_W 512

// Tiling: 128x16 pixels per 256-thread block (8 waves wave32, 8 px/thread)
#define TILE_W 128
#define TILE_H 16
#define ROWS   (TILE_H + 2)                   // 18 staged rows per channel
#define PITCH  132                            // 128 cols + 4 pad: rows 16B-aligned, +4 bank rotation
#define CH_STRIDE (ROWS * PITCH)              // 2376 floats per channel
#define INTERIOR_FLOATS (CH * CH_STRIDE)      // 7128
#define LHALO_OFF INTERIOR_FLOATS             // [3][18] left-halo column (x-1)
#define RHALO_OFF (INTERIOR_FLOATS + CH*ROWS) // [3][18] right-halo column (x+128)
#define LDS_FLOATS (INTERIOR_FLOATS + 2*CH*ROWS)  // 7236 floats = 28944 B

#define NROWSEGS (CH * ROWS)                  // 54 staged rows total
#define NHALO    (2 * CH * ROWS)              // 108 halo elements

typedef float v4f __attribute__((ext_vector_type(4)));

// CDNA5 async global->LDS copies (VGLOBAL encoding, ASYNCcnt-tracked).
__device__ __forceinline__ void async_ld_b128(uint32_t lds_byte_off, const float* gaddr) {
    asm volatile("global_load_async_to_lds_b128 %0, %1, off"
                 :: "v"(lds_byte_off), "v"(gaddr) : "memory");
}
__device__ __forceinline__ void async_ld_b32(uint32_t lds_byte_off, const float* gaddr) {
    asm volatile("global_load_async_to_lds_b32 %0, %1, off"
                 :: "v"(lds_byte_off), "v"(gaddr) : "memory");
}
__device__ __forceinline__ void wait_asynccnt0() {
    asm volatile("s_wait_asynccnt 0" ::: "memory");
}

// LDS read with halo-column redirect: ci in [-1, 128].
__device__ __forceinline__ float rd(const float* __restrict__ t, int c, int row, int ci) {
    int idx  = c * CH_STRIDE + row * PITCH + ci;
    int lidx = LHALO_OFF + c * ROWS + row;
    int ridx = RHALO_OFF + c * ROWS + row;
    idx = (ci < 0)          ? lidx : idx;
    idx = (ci > TILE_W - 1) ? ridx : idx;
    return t[idx];
}

__global__ __launch_bounds__(256) void sobel_mag_angle_kernel(
        const float* __restrict__ x,   // (B, 3, 512, 512)
        float* __restrict__ mag_out,   // (B, 512, 512)
        float* __restrict__ ang_out) { // (B, 512, 512)
    __shared__ float tile[LDS_FLOATS];

    const int tid = threadIdx.x;
    const int tx0 = blockIdx.x * TILE_W;
    const int ty0 = blockIdx.y * TILE_H;
    const int b   = blockIdx.z;

    // Low 32 bits of a generic pointer into LDS == byte offset in the LDS aperture.
    const uint32_t lds_base = (uint32_t)(uintptr_t)(&tile[0]);
    const float* xb = x + (size_t)b * CH * IMG_H * IMG_W;

    // ---- Stage interior: one staged row (128 cols) per 32-lane segment.
    // lxg fixed per lane; (ch,ly) depend only on tid>>5 + 8k (wave-uniform).
    const int lxg  = tid & 31;          // col group -> cols lxg*4 .. lxg*4+3
    const int rseg = tid >> 5;          // 0..7
#pragma unroll
    for (int k = 0; k < 7; ++k) {
        int rowIdx = rseg + 8 * k;                       // 0..55
        rowIdx = rowIdx < NROWSEGS ? rowIdx : NROWSEGS - 1;  // full EXEC; dup idempotent
        const int ch = rowIdx / ROWS;                    // 0..2
        const int ly = rowIdx - ch * ROWS;               // 0..17
        int gy = ty0 + ly - 1;
        gy = gy < 0 ? 0 : (gy > IMG_H - 1 ? IMG_H - 1 : gy);
        const float* src = xb + ((size_t)ch * IMG_H + gy) * IMG_W + (tx0 + lxg * 4);
        const uint32_t off = (uint32_t)(ch * CH_STRIDE + ly * PITCH + lxg * 4) * 4u;
        async_ld_b128(lds_base + off, src);
    }

    // ---- Stage halo columns (x-1 and x+128): 108 b32 loads, one full-EXEC sweep ----
    {
        int j = tid < NHALO ? tid : NHALO - 1;  // full EXEC; dup idempotent
        const int side = j & 1;                 // 0 = left (x-1), 1 = right (x+128)
        const int idx  = j >> 1;                // 0..53
        const int ch   = idx / ROWS;            // 0..2
        const int row  = idx - ch * ROWS;       // 0..17
        int gy = ty0 + row - 1;
        gy = gy < 0 ? 0 : (gy > IMG_H - 1 ? IMG_H - 1 : gy);
        int gx = side ? (tx0 + TILE_W) : (tx0 - 1);
        gx = gx < 0 ? 0 : (gx > IMG_W - 1 ? IMG_W - 1 : gx);
        const float* src = xb + ((size_t)ch * IMG_H + gy) * IMG_W + gx;
        const uint32_t off =
            (uint32_t)((side ? RHALO_OFF : LHALO_OFF) + ch * ROWS + row) * 4u;
        async_ld_b32(lds_base + off, src);
    }

    wait_asynccnt0();     // this wave's async copies landed in LDS
    __syncthreads();      // all waves' copies landed

    // ---- Compute: each thread -> 8 consecutive x pixels ----
    const int lxb = (tid & 15) * 8;  // 0..120 (x within tile)
    const int ly  = tid >> 4;        // 0..15  (y within tile)

    float bestm[8], gxs[8], gys[8];
#pragma unroll
    for (int j = 0; j < 8; ++j) { bestm[j] = -1.0f; gxs[j] = 0.0f; gys[j] = 0.0f; }

#pragma unroll
    for (int c = 0; c < CH; ++c) {
        // Separable Sobel: column sums over the 10-wide window.
        float sv[10], sh[10];
#pragma unroll
        for (int k = 0; k < 10; ++k) {
            const int ci = lxb - 1 + k;         // only k==0 / k==9 can leave [0,127]
            const float a0 = rd(tile, c, ly + 0, ci);
            const float a1 = rd(tile, c, ly + 1, ci);
            const float a2 = rd(tile, c, ly + 2, ci);
            sv[k] = (a0 + a2) + 2.0f * a1;      // vertical smoothing (for gx)
            sh[k] = a2 - a0;                    // vertical derivative (for gy)
        }
#pragma unroll
        for (int j = 0; j < 8; ++j) {
            const float gx = (sv[j + 2] - sv[j]) * 0.125f;
            const float gy = ((sh[j] + sh[j + 2]) + 2.0f * sh[j + 1]) * 0.125f;
            const float m = __builtin_amdgcn_sqrtf(gx * gx + gy * gy + 1e-9f);
            if (m > bestm[j]) {  // strict '>' => first-occurrence argmax (channel order)
                bestm[j] = m; gxs[j] = gx; gys[j] = gy;
            }
        }
    }

    v4f m4[2], a4[2];
#pragma unroll
    for (int j = 0; j < 8; ++j) {
        const float den = (gys[j] == 0.0f) ? 1e-9f : gys[j];
        float a = gxs[j] * __builtin_amdgcn_rcpf(den);   // clipped to +-10 below
        a = fminf(fmaxf(a, -10.0f), 10.0f);
        m4[j >> 2][j & 3] = bestm[j];  // == sqrt(gxs^2+gys^2+1e-9) of winning channel
        a4[j >> 2][j & 3] = a;
    }

    const int y = ty0 + ly;
    const int xcol = tx0 + lxb;
    const size_t o = ((size_t)b * IMG_H + y) * IMG_W + xcol;  // 32B aligned
    __builtin_nontemporal_store(m4[0], (v4f*)(mag_out + o));
    __builtin_nontemporal_store(m4[1], (v4f*)(mag_out + o + 4));
    __builtin_nontemporal_store(a4[0], (v4f*)(ang_out + o));
    __builtin_nontemporal_store(a4[1], (v4f*)(ang_out + o + 4));
}

extern "C" void kernel_launch(void* const* d_in, const int* in_sizes, int n_in,
                              void* d_out, int out_size, void* d_ws, size_t ws_size,
                              hipStream_t stream) {
    const float* x = (const float*)d_in[0];
    const int B = in_sizes[0] / (CH * IMG_H * IMG_W);  // 16

    float* out = (float*)d_out;
    float* mag = out;                                   // (B,512,512)
    float* ang = out + (size_t)B * IMG_H * IMG_W;       // (B,512,512)

    dim3 grid(IMG_W / TILE_W, IMG_H / TILE_H, B);       // 4 x 32 x 16 = 2048 blocks
    sobel_mag_angle_kernel<<<grid, 256, 0, stream>>>(x, mag, ang);
}